// PlaneRefineBatch_69741678952722
// MI455X (gfx1250) — compile-verified
//
#include <hip/hip_runtime.h>
#include <math.h>

#define UNDIST_ITERS 5
#define EPSF 1e-9f

typedef __attribute__((ext_vector_type(2))) float v2f;
typedef __attribute__((ext_vector_type(8))) float v8f;

struct Cam {
    float fx, fy, cx, cy, ifx, ify;
    float k1, k2, p1, p2, k3;
};

__device__ __forceinline__ float rcp_fast(float x) {
    return __builtin_amdgcn_rcpf(x);   // v_rcp_f32: ~1ulp, fine for iterative/self-correcting math
}

__device__ __forceinline__ void undistort(float u, float v, const Cam& c, float& xo, float& yo) {
    float x0 = (u - c.cx) * c.ifx;
    float y0 = (v - c.cy) * c.ify;
    float x = x0, y = y0;
#pragma unroll
    for (int i = 0; i < UNDIST_ITERS; ++i) {
        float r2 = x * x + y * y;
        float radial = 1.0f + r2 * (c.k1 + r2 * (c.k2 + r2 * c.k3));
        float dx = 2.0f * c.p1 * x * y + c.p2 * (r2 + 2.0f * x * x);
        float dy = c.p1 * (r2 + 2.0f * y * y) + 2.0f * c.p2 * x * y;
        float ir = rcp_fast(radial);
        x = (x0 - dx) * ir;
        y = (y0 - dy) * ir;
    }
    xo = x; yo = y;
}

__device__ __forceinline__ void project(float X, float Y, float Z, const Cam& c, float& u, float& v) {
    float z = (fabsf(Z) < 1e-6f) ? 1e-6f : Z;   // matches reference jnp.where semantics
    float iz = rcp_fast(z);
    float x = X * iz, y = Y * iz;
    float r2 = x * x + y * y;
    float radial = 1.0f + r2 * (c.k1 + r2 * (c.k2 + r2 * c.k3));
    float xd = x * radial + 2.0f * c.p1 * x * y + c.p2 * (r2 + 2.0f * x * x);
    float yd = y * radial + c.p1 * (r2 + 2.0f * y * y) + 2.0f * c.p2 * x * y;
    u = c.fx * xd + c.cx;
    v = c.fy * yd + c.cy;
}

// Process one point: triangulate + both reprojection errors. Returns 0.5*(e_cam+e_prj).
__device__ __forceinline__ float process_point(
    float cu, float cv, float pu, float pv,
    const float* Rm, const float* Tv, const Cam& cc, const Cam& pc,
    float& px, float& py, float& pz)
{
    float x1, y1, x2, y2;
    undistort(cu, cv, cc, x1, y1);
    undistort(pu, pv, pc, x2, y2);

    // Projector DLT rows: m3 = x2*R2 - R0 ; m4 = y2*R2 - R1
    float m3x = x2 * Rm[6] - Rm[0];
    float m3y = x2 * Rm[7] - Rm[1];
    float m3z = x2 * Rm[8] - Rm[2];
    float m4x = y2 * Rm[6] - Rm[3];
    float m4y = y2 * Rm[7] - Rm[4];
    float m4z = y2 * Rm[8] - Rm[5];
    float b3 = x2 * Tv[2] - Tv[0];
    float b4 = y2 * Tv[2] - Tv[1];

    // Normal equations AtA (symmetric) and Atb; camera rows contribute analytically.
    float a00 = 1.0f + m3x * m3x + m4x * m4x + EPSF;
    float a01 = m3x * m3y + m4x * m4y;
    float a02 = -x1 + m3x * m3z + m4x * m4z;
    float a11 = 1.0f + m3y * m3y + m4y * m4y + EPSF;
    float a12 = -y1 + m3y * m3z + m4y * m4z;
    float a22 = x1 * x1 + y1 * y1 + m3z * m3z + m4z * m4z + EPSF;
    float bx = m3x * b3 + m4x * b4;
    float by = m3y * b3 + m4y * b4;
    float bz = m3z * b3 + m4z * b4;

    // Symmetric 3x3 inverse via cofactors; pts = -A^{-1} b  (precise divide here)
    float c00 = a11 * a22 - a12 * a12;
    float c01 = a02 * a12 - a01 * a22;
    float c02 = a01 * a12 - a02 * a11;
    float c11 = a00 * a22 - a02 * a02;
    float c12 = a01 * a02 - a00 * a12;
    float c22 = a00 * a11 - a01 * a01;
    float det = a00 * c00 + a01 * c01 + a02 * c02;
    float idet = -1.0f / det;
    px = (c00 * bx + c01 * by + c02 * bz) * idet;
    py = (c01 * bx + c11 * by + c12 * bz) * idet;
    pz = (c02 * bx + c12 * by + c22 * bz) * idet;

    // Reprojection errors
    float uc, vc;
    project(px, py, pz, cc, uc, vc);
    float qx = Rm[0] * px + Rm[1] * py + Rm[2] * pz + Tv[0];
    float qy = Rm[3] * px + Rm[4] * py + Rm[5] * pz + Tv[1];
    float qz = Rm[6] * px + Rm[7] * py + Rm[8] * pz + Tv[2];
    float up, vp;
    project(qx, qy, qz, pc, up, vp);
    float du = uc - cu, dv = vc - cv;
    float e_cam = sqrtf(du * du + dv * dv + EPSF);
    du = up - pu; dv = vp - pv;
    float e_prj = sqrtf(du * du + dv * dv + EPSF);
    return 0.5f * (e_cam + e_prj);
}

__global__ void __launch_bounds__(256)
triangulate_kernel(const float* __restrict__ camPts, const float* __restrict__ prjPts,
                   const float* __restrict__ Rg, const float* __restrict__ Tg,
                   const float* __restrict__ camKg, const float* __restrict__ camDg,
                   const float* __restrict__ prjKg, const float* __restrict__ prjDg,
                   float* __restrict__ ptsOut, double* __restrict__ acc, int P)
{
    // Stage 40 parameter floats through LDS once per block.
    __shared__ float sp[40];
    int tid = threadIdx.x;
    if (tid < 40) {
        float v;
        if (tid < 9)       v = Rg[tid];
        else if (tid < 12) v = Tg[tid - 9];
        else if (tid < 21) v = camKg[tid - 12];
        else if (tid < 26) v = camDg[tid - 21];
        else if (tid < 35) v = prjKg[tid - 26];
        else               v = prjDg[tid - 35];
        sp[tid] = v;
    }
    __syncthreads();

    Cam cc, pc;
    cc.fx = sp[12]; cc.cx = sp[14]; cc.fy = sp[16]; cc.cy = sp[17];
    cc.ifx = 1.0f / cc.fx; cc.ify = 1.0f / cc.fy;
    cc.k1 = sp[21]; cc.k2 = sp[22]; cc.p1 = sp[23]; cc.p2 = sp[24]; cc.k3 = sp[25];
    pc.fx = sp[26]; pc.cx = sp[28]; pc.fy = sp[30]; pc.cy = sp[31];
    pc.ifx = 1.0f / pc.fx; pc.ify = 1.0f / pc.fy;
    pc.k1 = sp[35]; pc.k2 = sp[36]; pc.p1 = sp[37]; pc.p2 = sp[38]; pc.k3 = sp[39];
    float Rm[9], Tv[3];
#pragma unroll
    for (int i = 0; i < 9; ++i) Rm[i] = sp[i];
#pragma unroll
    for (int i = 0; i < 3; ++i) Tv[i] = sp[9 + i];

    int t = blockIdx.x * blockDim.x + threadIdx.x;
    int p0 = 4 * t;
    float esum = 0.0f;

    if (p0 + 3 < P) {
        // Vector path: 4 consecutive points -> b128 loads/stores.
        const float4* cam4 = (const float4*)camPts;
        const float4* prj4 = (const float4*)prjPts;
        float4 c0 = cam4[2 * t], c1 = cam4[2 * t + 1];
        float4 q0 = prj4[2 * t], q1 = prj4[2 * t + 1];
        float cu[4] = {c0.x, c0.z, c1.x, c1.z};
        float cv[4] = {c0.y, c0.w, c1.y, c1.w};
        float pu[4] = {q0.x, q0.z, q1.x, q1.z};
        float pv[4] = {q0.y, q0.w, q1.y, q1.w};
        float o[12];
#pragma unroll
        for (int j = 0; j < 4; ++j) {
            float px, py, pz;
            esum += process_point(cu[j], cv[j], pu[j], pv[j], Rm, Tv, cc, pc, px, py, pz);
            o[3 * j + 0] = px; o[3 * j + 1] = py; o[3 * j + 2] = pz;
        }
        float4* out4 = (float4*)ptsOut;
        out4[3 * t + 0] = make_float4(o[0], o[1], o[2], o[3]);
        out4[3 * t + 1] = make_float4(o[4], o[5], o[6], o[7]);
        out4[3 * t + 2] = make_float4(o[8], o[9], o[10], o[11]);
    } else {
        // Scalar tail (not taken for the given sizes).
        for (int j = 0; j < 4; ++j) {
            int p = p0 + j;
            if (p < P) {
                float px, py, pz;
                esum += process_point(camPts[2 * p], camPts[2 * p + 1],
                                      prjPts[2 * p], prjPts[2 * p + 1],
                                      Rm, Tv, cc, pc, px, py, pz);
                ptsOut[3 * p + 0] = px;
                ptsOut[3 * p + 1] = py;
                ptsOut[3 * p + 2] = pz;
            }
        }
    }

    // Wave-level reduction of esum. EXEC is all-ones here (whole wave reconverged).
#if __has_builtin(__builtin_amdgcn_wmma_f32_16x16x4_f32)
    // A(16x4): lane L contributes A[L%16][*] = e[L] twice each half -> row-sum = 2(e[m]+e[m+16]).
    // B(4x16) = ones. D[m][n] = 2(e[m]+e[m+16]); lane's 8 acc VGPRs cover m=0..7 (lanes 0-15)
    // or m=8..15 (lanes 16-31). Sum*0.5 + shfl_xor(16) = exact f32 sum over 32 lanes.
    v2f A; A.x = esum; A.y = esum;
    v2f Bo; Bo.x = 1.0f; Bo.y = 1.0f;
    v8f Cz = {};
    v8f D = __builtin_amdgcn_wmma_f32_16x16x4_f32(false, A, false, Bo, (short)0, Cz, false, false);
    float half_sum = (D[0] + D[1] + D[2] + D[3] + D[4] + D[5] + D[6] + D[7]) * 0.5f;
    float wsum = half_sum + __shfl_xor(half_sum, 16, 32);
#else
    float wsum = esum;
    for (int off = 16; off > 0; off >>= 1) wsum += __shfl_xor(wsum, off, 32);
#endif
    if ((threadIdx.x & 31) == 0) {
        atomicAdd(acc, (double)wsum);   // global_atomic_add_f64
    }
}

__global__ void zero_acc_kernel(double* acc) {
    if (threadIdx.x == 0 && blockIdx.x == 0) *acc = 0.0;
}

__global__ void finalize_kernel(const double* __restrict__ acc, float* __restrict__ errOut, int P) {
    if (threadIdx.x == 0 && blockIdx.x == 0) {
        *errOut = (float)(*acc / (double)P);
    }
}

extern "C" void kernel_launch(void* const* d_in, const int* in_sizes, int n_in,
                              void* d_out, int out_size, void* d_ws, size_t ws_size,
                              hipStream_t stream) {
    const float* camPts = (const float*)d_in[0];
    const float* prjPts = (const float*)d_in[1];
    const float* R      = (const float*)d_in[2];
    const float* T      = (const float*)d_in[3];
    const float* camK   = (const float*)d_in[4];
    const float* camD   = (const float*)d_in[5];
    const float* prjK   = (const float*)d_in[6];
    const float* prjD   = (const float*)d_in[7];
    float*  out = (float*)d_out;
    double* acc = (double*)d_ws;

    const int P = in_sizes[0] / 2;            // total points B*N
    const int nthreads = (P + 3) / 4;         // 4 points per thread
    const int block = 256;                    // 8 wave32 waves per workgroup
    const int grid = (nthreads + block - 1) / block;

    zero_acc_kernel<<<1, 1, 0, stream>>>(acc);
    triangulate_kernel<<<grid, block, 0, stream>>>(camPts, prjPts, R, T, camK, camD,
                                                   prjK, prjD, out, acc, P);
    finalize_kernel<<<1, 1, 0, stream>>>(acc, out + (size_t)3 * P, P);
}